// MACUnit_13125420057186
// MI455X (gfx1250) — compile-verified
//
#include <hip/hip_runtime.h>

#define B_ 128
#define S_ 128
#define N_ 196
#define D_ 512
#define T_ 12

typedef float v2f __attribute__((ext_vector_type(2)));
typedef float v4f __attribute__((ext_vector_type(4)));
typedef float v8f __attribute__((ext_vector_type(8)));

static __device__ __forceinline__ v2f mk2(float x, float y) {
    v2f r; r.x = x; r.y = y; return r;
}

// ---------------------------------------------------------------------------
// One K-partition of a 16x16 output tile, fp32 WMMA 16x16x4.
// Within each 16-wide K macro-step we permute K so that pair-p lanes feed
// global K = kw + 8p + {0..7}: each lane then reads 8 contiguous floats of A
// (and of B when BTRANS) -> two global_load_b128 per matrix per 4 WMMAs.
// The same permutation is applied to A and B, so the K-sum is unchanged.
// ---------------------------------------------------------------------------
template <bool BTRANS>
static __device__ __forceinline__ void gemm_part(
    v8f& acc, const float* __restrict__ A, int lda,
    const float* __restrict__ Bm, int ldb,
    int m, int n, int pair, int kOff, int Kpart)   // Kpart % 16 == 0
{
    const float* ap = A + (long)m * lda + pair * 8;
    const float* bp = BTRANS ? (Bm + (long)n * ldb + kOff + pair * 8)
                             : (Bm + ((long)kOff + pair * 8) * ldb + n);
    for (int ks = 0; ks < Kpart; ks += 16) {
        const v4f a01 = *(const v4f*)ap;
        const v4f a23 = *(const v4f*)(ap + 4);
        v4f b01, b23;
        if (BTRANS) {
            b01 = *(const v4f*)bp;
            b23 = *(const v4f*)(bp + 4);
        } else {
            b01.x = bp[0L * ldb]; b01.y = bp[1L * ldb];
            b01.z = bp[2L * ldb]; b01.w = bp[3L * ldb];
            b23.x = bp[4L * ldb]; b23.y = bp[5L * ldb];
            b23.z = bp[6L * ldb]; b23.w = bp[7L * ldb];
        }
        acc = __builtin_amdgcn_wmma_f32_16x16x4_f32(
            false, mk2(a01.x, a01.y), false, mk2(b01.x, b01.y), (short)0, acc, false, false);
        acc = __builtin_amdgcn_wmma_f32_16x16x4_f32(
            false, mk2(a01.z, a01.w), false, mk2(b01.z, b01.w), (short)0, acc, false, false);
        acc = __builtin_amdgcn_wmma_f32_16x16x4_f32(
            false, mk2(a23.x, a23.y), false, mk2(b23.x, b23.y), (short)0, acc, false, false);
        acc = __builtin_amdgcn_wmma_f32_16x16x4_f32(
            false, mk2(a23.z, a23.w), false, mk2(b23.z, b23.w), (short)0, acc, false, false);
        ap += 16;
        bp += BTRANS ? 16 : 16L * ldb;
    }
}

// ---------------------------------------------------------------------------
// fp32 WMMA GEMM:  out[M,N] = concat(A0,A1)[M,K] * B  (+bias)(*scale)
//   BTRANS=1: B is [N,K] row-major ("X @ W.T");  BTRANS=0: B is [K,N].
//   blockDim=256 (8 waves); each wave owns one 16x16 C tile.
//   grid = (N/128, M/16, Z); Z batches B/bias/out by given strides.
// ---------------------------------------------------------------------------
template <bool BTRANS>
__global__ __launch_bounds__(256) void wmma_gemm_f32(
    const float* __restrict__ A0, const float* __restrict__ A1,
    int lda0, int lda1, int Ksplit, int K,
    const float* __restrict__ Bm, int ldb,
    const float* __restrict__ bias, const float* __restrict__ scale,
    float* __restrict__ out, int ldc,
    long strideB, long strideBias, long strideOut)
{
    const int z = blockIdx.z;
    Bm += (long)z * strideB;
    if (bias) bias += (long)z * strideBias;
    out += (long)z * strideOut;

    const int lane  = threadIdx.x & 31;
    const int wave  = threadIdx.x >> 5;
    const int row16 = lane & 15;
    const int pair  = lane >> 4;              // 0 or 1
    const int ntile = blockIdx.x * 8 + wave;
    const int mtile = blockIdx.y;
    const int m = mtile * 16 + row16;         // A-fragment row for this lane
    const int n = ntile * 16 + row16;         // B-fragment col for this lane

    v8f acc = {};
    gemm_part<BTRANS>(acc, A0, lda0, Bm, ldb, m, n, pair, 0, Ksplit);
    if (Ksplit < K)   // scalar (kernarg) condition -> uniform, EXEC stays full
        gemm_part<BTRANS>(acc, A1, lda1, Bm, ldb, m, n, pair, Ksplit, K - Ksplit);

    const float bb = bias  ? bias[n]  : 0.0f;
    const float sc = scale ? scale[n] : 1.0f;
    const int mbase = mtile * 16 + pair * 8;  // C layout: vgpr j -> M=j / j+8
    #pragma unroll
    for (int j = 0; j < 8; ++j) {
        out[(long)(mbase + j) * ldc + n] = (acc[j] + bb) * sc;
    }
}

// ---------------------------------------------------------------------------
// Control-unit attention over ctx:  logit[s] = ctx[b,s,:]·g[b]; softmax over s;
// c[b,:] = sum_s aw[s]*ctx[b,s,:];  also writes v = c * Wra (read-unit input).
// One block per batch, 256 threads.
// ---------------------------------------------------------------------------
__global__ __launch_bounds__(256) void ctx_attn(
    const float* __restrict__ ctx, const float* __restrict__ g,
    const float* __restrict__ Wra,
    float* __restrict__ c_out, float* __restrict__ v_out)
{
    __shared__ float g_s[D_];
    __shared__ float l_s[S_];
    __shared__ float red[8];
    const int b = blockIdx.x, tid = threadIdx.x;
    const int lane = tid & 31, wave = tid >> 5;

    for (int d = tid; d < D_; d += 256) g_s[d] = g[b * D_ + d];
    __syncthreads();

    const float* cb = ctx + (long)b * S_ * D_;
    for (int s = wave; s < S_; s += 8) {
        float acc = 0.f;
        const float* row = cb + (long)s * D_;
        for (int d = lane; d < D_; d += 32) acc += row[d] * g_s[d];
        for (int off = 16; off > 0; off >>= 1) acc += __shfl_xor(acc, off, 32);
        if (lane == 0) l_s[s] = acc;
    }
    __syncthreads();

    // block softmax over S_ (=128) entries
    float val = (tid < S_) ? l_s[tid] : -INFINITY;
    float mx = val;
    for (int off = 16; off > 0; off >>= 1) mx = fmaxf(mx, __shfl_xor(mx, off, 32));
    if (lane == 0) red[wave] = mx;
    __syncthreads();
    mx = red[0];
    for (int w = 1; w < 8; ++w) mx = fmaxf(mx, red[w]);
    const float e = (tid < S_) ? __expf(val - mx) : 0.f;
    float sm = e;
    for (int off = 16; off > 0; off >>= 1) sm += __shfl_xor(sm, off, 32);
    __syncthreads();
    if (lane == 0) red[wave] = sm;
    __syncthreads();
    sm = 0.f;
    for (int w = 0; w < 8; ++w) sm += red[w];
    if (tid < S_) l_s[tid] = e / sm;
    __syncthreads();

    for (int d = tid; d < D_; d += 256) {
        float acc = 0.f;
        #pragma unroll 4
        for (int s = 0; s < S_; ++s) acc += l_s[s] * cb[(long)s * D_ + d];
        c_out[b * D_ + d] = acc;
        v_out[b * D_ + d] = acc * Wra[d];     // v = c * Wra for the u-GEMM
    }
}

// ---------------------------------------------------------------------------
// Read-unit attention over k:  w[d] = mm[b,d]*u[b,d] + u[b,D+d];
// logit[n] = sum_d k[b,d,n]*w[d]; softmax over n; r[b,d] = sum_n rw[n]*k[b,d,n].
// One block per batch, 256 threads.
// ---------------------------------------------------------------------------
__global__ __launch_bounds__(256) void k_attn(
    const float* __restrict__ kmat, const float* __restrict__ mm,
    const float* __restrict__ u, float* __restrict__ r)
{
    __shared__ float w_s[D_];
    __shared__ float l_s[256];
    __shared__ float red[8];
    const int b = blockIdx.x, tid = threadIdx.x;
    const int lane = tid & 31, wave = tid >> 5;

    const float* ub = u + (long)b * 2 * D_;
    for (int d = tid; d < D_; d += 256)
        w_s[d] = mm[b * D_ + d] * ub[d] + ub[D_ + d];
    __syncthreads();

    const float* kb = kmat + (long)b * D_ * N_;
    float logit = -INFINITY;
    if (tid < N_) {
        float acc = 0.f;
        #pragma unroll 4
        for (int d = 0; d < D_; ++d) acc += kb[(long)d * N_ + tid] * w_s[d];
        logit = acc;
    }
    // block softmax over 256 slots (pads contribute exp(-inf)=0)
    float mx = logit;
    for (int off = 16; off > 0; off >>= 1) mx = fmaxf(mx, __shfl_xor(mx, off, 32));
    if (lane == 0) red[wave] = mx;
    __syncthreads();
    mx = red[0];
    for (int w = 1; w < 8; ++w) mx = fmaxf(mx, red[w]);
    const float e = (tid < N_) ? __expf(logit - mx) : 0.f;
    float sm = e;
    for (int off = 16; off > 0; off >>= 1) sm += __shfl_xor(sm, off, 32);
    __syncthreads();
    if (lane == 0) red[wave] = sm;
    __syncthreads();
    sm = 0.f;
    for (int w = 0; w < 8; ++w) sm += red[w];
    l_s[tid] = e / sm;
    __syncthreads();

    for (int d = wave; d < D_; d += 8) {
        float acc = 0.f;
        const float* row = kb + (long)d * N_;
        for (int n = lane; n < N_; n += 32) acc += l_s[n] * row[n];
        for (int off = 16; off > 0; off >>= 1) acc += __shfl_xor(acc, off, 32);
        if (lane == 0) r[b * D_ + d] = acc;
    }
}

// Broadcast initial states: c = ctrl0, m = mem0 over the batch.
__global__ void init_cm(const float* __restrict__ ctrl0,
                        const float* __restrict__ mem0,
                        float* __restrict__ c, float* __restrict__ m)
{
    const int i = blockIdx.x * blockDim.x + threadIdx.x;   // B_*D_ threads
    const int d = i & (D_ - 1);
    c[i] = ctrl0[d];
    m[i] = mem0[d];
}

// ---------------------------------------------------------------------------
extern "C" void kernel_launch(void* const* d_in, const int* in_sizes, int n_in,
                              void* d_out, int out_size, void* d_ws, size_t ws_size,
                              hipStream_t stream) {
    (void)in_sizes; (void)n_in; (void)out_size; (void)ws_size;
    const float* ctx   = (const float*)d_in[0];
    const float* q     = (const float*)d_in[1];
    const float* kmat  = (const float*)d_in[2];
    const float* mem0  = (const float*)d_in[3];
    const float* ctrl0 = (const float*)d_in[4];
    const float* Wp    = (const float*)d_in[5];
    const float* bp    = (const float*)d_in[6];
    const float* Wcq   = (const float*)d_in[7];
    const float* bcq   = (const float*)d_in[8];
    const float* Wca   = (const float*)d_in[9];
    // d_in[10] = bca : constant across softmax axis -> drops out
    const float* Wrm   = (const float*)d_in[11];
    const float* brm   = (const float*)d_in[12];
    const float* Wrc   = (const float*)d_in[13];
    // d_in[14] = brc : per-batch constant -> drops out of softmax
    const float* Wra   = (const float*)d_in[15];
    // d_in[16] = bra : constant -> drops out
    const float* Wwc   = (const float*)d_in[17];
    const float* bwc   = (const float*)d_in[18];
    float* out = (float*)d_out;

    // workspace layout (floats); total ~5.3 MB
    float* ws = (float*)d_ws;
    const long BD = (long)B_ * D_;
    float* P    = ws;                 // [T, B, D] precomputed p(t)
    float* cbuf = P + (long)T_ * BD;  // [B, D]
    float* g    = cbuf + BD;          // [B, D]  (cc * Wca)
    float* v    = g + BD;             // [B, D]  (c * Wra)
    float* mmb  = v + BD;             // [B, D]
    float* u    = mmb + BD;           // [B, 2D]
    float* rbuf = u + 2 * BD;         // [B, D]
    float* m0   = rbuf + BD;          // [B, D]
    float* m1   = m0 + BD;            // [B, D]

    const dim3 blk(256);

    // init recurrent state
    init_cm<<<dim3((B_ * D_) / 256), blk, 0, stream>>>(ctrl0, mem0, cbuf, m0);

    // precompute P[t] = q @ Wp[t].T + bp[t] for all t (batched over grid.z)
    wmma_gemm_f32<true><<<dim3(D_ / 128, B_ / 16, T_), blk, 0, stream>>>(
        q, nullptr, 2 * D_, 0, 2 * D_, 2 * D_,
        Wp, 2 * D_, bp, nullptr, P, D_,
        (long)D_ * 2 * D_, (long)D_, BD);

    float* mcur = m0;
    float* mnxt = m1;
    for (int t = 0; t < T_; ++t) {
        // g = (concat(c, P[t]) @ Wcq.T + bcq) * Wca
        wmma_gemm_f32<true><<<dim3(D_ / 128, B_ / 16, 1), blk, 0, stream>>>(
            cbuf, P + (long)t * BD, D_, D_, D_, 2 * D_,
            Wcq, 2 * D_, bcq, Wca, g, D_, 0, 0, 0);
        // softmax attention over ctx -> new c, and v = c * Wra
        ctx_attn<<<dim3(B_), blk, 0, stream>>>(ctx, g, Wra, cbuf, v);
        // mm = m @ Wrm.T + brm
        wmma_gemm_f32<true><<<dim3(D_ / 128, B_ / 16, 1), blk, 0, stream>>>(
            mcur, nullptr, D_, 0, D_, D_,
            Wrm, D_, brm, nullptr, mmb, D_, 0, 0, 0);
        // u = v @ Wrc   (Wrc row-major [D, 2D]; bias const drops in softmax)
        wmma_gemm_f32<false><<<dim3(2 * D_ / 128, B_ / 16, 1), blk, 0, stream>>>(
            v, nullptr, D_, 0, D_, D_,
            Wrc, 2 * D_, nullptr, nullptr, u, 2 * D_, 0, 0, 0);
        // read attention over k -> r
        k_attn<<<dim3(B_), blk, 0, stream>>>(kmat, mmb, u, rbuf);
        // m' = concat(r, m) @ Wwc.T + bwc   (last step writes d_out directly)
        float* mdst = (t == T_ - 1) ? out : mnxt;
        wmma_gemm_f32<true><<<dim3(D_ / 128, B_ / 16, 1), blk, 0, stream>>>(
            rbuf, mcur, D_, D_, D_, 2 * D_,
            Wwc, 2 * D_, bwc, nullptr, mdst, D_, 0, 0, 0);
        float* tmp = mcur; mcur = mnxt; mnxt = tmp;
    }
}